// CTCLoss_24326694764986
// MI455X (gfx1250) — compile-verified
//
#include <hip/hip_runtime.h>

// CTC loss forward (T=1024, B=128, C=96, S=200) for gfx1250.
// One workgroup per batch element; L=401 alpha states on 416 threads (13 wave32s).
// Alpha double-buffered in LDS; per-timestep emission slice (96 floats) staged
// into LDS via the Tensor Data Mover (tensor_load_to_lds + s_wait_tensorcnt),
// double-buffered so the TDM DMA overlaps the logsumexp recursion.

#define NEGV (-1e30f)

constexpr int T_ = 1024;
constexpr int B_ = 128;
constexpr int C_ = 96;
constexpr int S_ = 200;
constexpr int L_ = 2 * S_ + 1;        // 401 extended states
constexpr int NTHREADS = 416;         // 13 waves of 32, covers L_=401

#if __has_builtin(__builtin_amdgcn_tensor_load_to_lds) && \
    __has_builtin(__builtin_amdgcn_s_wait_tensorcnt)
#define CTC_USE_TDM 1
#endif

typedef unsigned int v4u __attribute__((ext_vector_type(4)));
typedef int          v8i __attribute__((ext_vector_type(8)));
typedef int          v4i __attribute__((ext_vector_type(4)));

// Stage 96 contiguous floats (384 B) from global into LDS.
// Executed by wave 0 only. TDM path: 1-D tensor descriptor per ISA ch.8.
__device__ __forceinline__ void stage_lp(const float* __restrict__ gsrc,
                                         float* __restrict__ ldsdst) {
#ifdef CTC_USE_TDM
  unsigned long long ga = (unsigned long long)(uintptr_t)gsrc;
  unsigned int lds_off  = (unsigned int)(uintptr_t)ldsdst;  // low 32 bits = LDS byte addr

  // D# group 0 (128b): [1:0]=count=1, [63:32]=lds_addr, [120:64]=global_addr,
  //                    [127:126]=type=2
  v4u g0;
  g0[0] = 1u;
  g0[1] = lds_off;
  g0[2] = (unsigned int)(ga & 0xFFFFFFFFull);
  g0[3] = ((unsigned int)(ga >> 32) & 0x01FFFFFFu) | (2u << 30);

  // D# group 1 (256b): workgroup_mask=0, data_size=2 (4 bytes), no pad/iterate,
  // tensor_dim0=96 (bits 79:48), tensor_dim1=1 (bits 111:80),
  // tile_dim0=96 (bits 127:112), tile_dim1/2=0, tensor_dim0_stride=96 (bits 207:160).
  v8i g1;
  g1[0] = (int)(2u << 16);      // data_size = 4 bytes
  g1[1] = (int)(96u << 16);     // tensor_dim0[15:0] in [31:16]
  g1[2] = (int)(1u << 16);      // tensor_dim0[31:16]=0 | tensor_dim1[15:0]=1
  g1[3] = (int)(96u << 16);     // tensor_dim1[31:16]=0 | tile_dim0=96
  g1[4] = 0;                    // tile_dim1=0 (unused), tile_dim2=0
  g1[5] = 96;                   // tensor_dim0_stride[31:0]
  g1[6] = 0;                    // stride hi bits / tensor_dim1_stride lo
  g1[7] = 0;

  v4i gz4 = {0, 0, 0, 0};               // groups 2/3 unused (<=2-D tensor)
  v8i gz8 = {0, 0, 0, 0, 0, 0, 0, 0};   // extra descriptor group (clang-23 form)
  __builtin_amdgcn_tensor_load_to_lds(g0, g1, gz4, gz4, gz8, 0);
#else
  int lane = (int)(threadIdx.x & 31);
#pragma unroll
  for (int k = 0; k < C_ / 32; ++k)
    ldsdst[lane + 32 * k] = gsrc[lane + 32 * k];
#endif
}

__device__ __forceinline__ void wait_stage() {
#ifdef CTC_USE_TDM
  __builtin_amdgcn_s_wait_tensorcnt(0);
#endif
}

__global__ __launch_bounds__(NTHREADS) void ctc_alpha_kernel(
    const float* __restrict__ log_probs,     // [T, B, C]
    const int*   __restrict__ targets,       // [B, S]
    const int*   __restrict__ input_lengths, // [B]
    const int*   __restrict__ target_lengths,// [B]
    float*       __restrict__ per_batch)     // [B] loss/target_len
{
  __shared__ float abuf[2][L_ + 15];   // alpha double buffer (padded)
  __shared__ float lp[2][128];         // emission slice double buffer (96 used)

  const int b  = (int)blockIdx.x;
  const int s  = (int)threadIdx.x;
  const int il = input_lengths[b];
  const int tl = target_lengths[b];
  const bool active = (s < L_);
  const bool issuer = (threadIdx.x < 32);  // wave 0 drives TDM

  // Extended label and skip-allow flag, resident in registers per thread.
  int  ext_s = 0;
  bool allow = false;
  if (active && (s & 1)) {
    ext_s = targets[b * S_ + (s >> 1)];                       // labels in [1, C-1]
    if (s >= 3)
      allow = (ext_s != targets[b * S_ + (s >> 1) - 1]);      // ext[s] != ext[s-2]
  }

  const float* lpbase = log_probs + (size_t)b * C_;           // time stride = B_*C_

  // Prime lp[0] (t=0) and lp[1] (t=1).
  if (issuer) {
    stage_lp(lpbase, &lp[0][0]);
    if (il > 1) stage_lp(lpbase + (size_t)(B_ * C_), &lp[1][0]);
    wait_stage();
  }
  __syncthreads();

  // alpha_0
  if (active) {
    float e0 = lp[0][ext_s];
    abuf[0][s] = (s <= 1) ? e0 : NEGV;
  }
  __syncthreads();

  // Recurrence over time; early exit at t = il-1 (alphas beyond are unused).
  int cur = 0;
  for (int t = 1; t < il; ++t) {
    const int nxt = t & 1;
    // Prefetch lp for t+1 into the buffer that held lp[t-1] (free after last barrier).
    if (issuer && (t + 1 < il))
      stage_lp(lpbase + (size_t)(t + 1) * (B_ * C_), &lp[(t + 1) & 1][0]);

    if (active) {
      float a0 = abuf[cur][s];
      float a1 = (s >= 1) ? abuf[cur][s - 1] : NEGV;
      float a2 = allow    ? abuf[cur][s - 2] : NEGV;
      float m  = fmaxf(a0, fmaxf(a1, a2));
      float lse = m + __logf(__expf(a0 - m) + __expf(a1 - m) + __expf(a2 - m));
      abuf[nxt][s] = lse + lp[t & 1][ext_s];
    }

    if (issuer) wait_stage();   // lp[t+1] committed to LDS before next step
    __syncthreads();
    cur = nxt;
  }

  // Final 2-state logsumexp at t = il-1 (alpha lives in abuf[cur]).
  if (threadIdx.x == 0) {
    float a1 = abuf[cur][2 * tl - 1];
    float a2 = abuf[cur][2 * tl];
    float m  = fmaxf(a1, a2);
    float ll = m + __logf(__expf(a1 - m) + __expf(a2 - m));
    float loss = -ll;
    if (loss > 1e29f) loss = 0.0f;                 // zero_infinity
    per_batch[b] = loss / (float)tl;               // reduction pre-mean
  }
}

// Deterministic tree reduction over the 128 per-batch values -> mean.
__global__ __launch_bounds__(128) void ctc_reduce_mean(
    const float* __restrict__ per_batch, float* __restrict__ out)
{
  __shared__ float sh[128];
  int i = (int)threadIdx.x;
  sh[i] = per_batch[i];
  __syncthreads();
  for (int off = 64; off > 0; off >>= 1) {
    if (i < off) sh[i] += sh[i + off];
    __syncthreads();
  }
  if (i == 0) out[0] = sh[0] * (1.0f / (float)B_);
}

extern "C" void kernel_launch(void* const* d_in, const int* in_sizes, int n_in,
                              void* d_out, int out_size, void* d_ws, size_t ws_size,
                              hipStream_t stream) {
  const float* log_probs      = (const float*)d_in[0];
  const int*   targets        = (const int*)d_in[1];
  const int*   input_lengths  = (const int*)d_in[2];
  const int*   target_lengths = (const int*)d_in[3];
  float*       per_batch      = (float*)d_ws;   // 128 floats of scratch

  ctc_alpha_kernel<<<B_, NTHREADS, 0, stream>>>(
      log_probs, targets, input_lengths, target_lengths, per_batch);
  ctc_reduce_mean<<<1, 128, 0, stream>>>(per_batch, (float*)d_out);
}